// AlphaCompositor_73521250173218
// MI455X (gfx1250) — compile-verified
//
#include <hip/hip_runtime.h>
#include <stdint.h>

// Problem shape (from reference): fragments/alphas (N,K,H,W), features (C,P)
#define N_    8
#define K_    8
#define H_    256
#define W_    256
#define C_    32
#define HW_   (H_ * W_)
#define TILE_ 256          // pixels handled per workgroup

typedef unsigned int u32x4 __attribute__((ext_vector_type(4)));
typedef int          i32x4 __attribute__((ext_vector_type(4)));
typedef int          i32x8 __attribute__((ext_vector_type(8)));

#if defined(__has_builtin)
#if __has_builtin(__builtin_amdgcn_tensor_load_to_lds) && \
    __has_builtin(__builtin_amdgcn_s_wait_tensorcnt)
#define HAVE_TDM 1
#endif
#endif
#ifndef HAVE_TDM
#define HAVE_TDM 0
#endif

#if HAVE_TDM
// Build a 2D Tensor-DMA descriptor (D#) and issue TENSOR_LOAD_TO_LDS.
// Layout per CDNA5 ISA ch.8:
//   group0: [1:0]=count(1), [63:32]=lds_addr, [120:64]=global_addr, [127:126]=type(2)
//   group1: [17:16]=data_size, [79:48]=tensor_dim0, [111:80]=tensor_dim1,
//           [127:112]=tile_dim0, [143:128]=tile_dim1, [207:160]=tensor_dim0_stride
__device__ __forceinline__ void tdm_load_2d(uint32_t lds_addr, uint64_t gaddr,
                                            uint32_t data_size_code,
                                            uint32_t tile_d0, uint32_t tile_d1,
                                            uint64_t tensor_d0, uint64_t tensor_d1,
                                            uint64_t stride0)
{
    u32x4 g0;
    g0[0] = 1u;                                                    // count=1, user D#
    g0[1] = lds_addr;                                              // LDS byte address
    g0[2] = (uint32_t)(gaddr & 0xFFFFFFFFull);                     // global_addr[31:0]
    g0[3] = ((uint32_t)(gaddr >> 32) & 0x01FFFFFFu) | 0x80000000u; // [56:32] | type=2

    i32x8 g1;
    g1[0] = (int)((data_size_code & 3u) << 16);                    // wg_mask=0, data_size
    g1[1] = (int)((tensor_d0 & 0xFFFFull) << 16);                  // tensor_dim0[15:0]
    g1[2] = (int)(((tensor_d0 >> 16) & 0xFFFFull) |
                  ((tensor_d1 & 0xFFFFull) << 16));                // td0[31:16] | td1[15:0]
    g1[3] = (int)(((tensor_d1 >> 16) & 0xFFFFull) |
                  ((uint64_t)(tile_d0 & 0xFFFFu) << 16));          // td1[31:16] | tile_dim0
    g1[4] = (int)(tile_d1 & 0xFFFFu);                              // tile_dim1, tile_dim2=0
    g1[5] = (int)(stride0 & 0xFFFFFFFFull);                        // dim0_stride[31:0]
    g1[6] = (int)((stride0 >> 32) & 0xFFFFull);                    // dim0_stride[47:32]
    g1[7] = 0;                                                     // dim1_stride unused (2D)

    i32x4 z4 = {0, 0, 0, 0};
#if defined(__clang_major__) && (__clang_major__ >= 23)
    i32x8 z8 = {0, 0, 0, 0, 0, 0, 0, 0};
    __builtin_amdgcn_tensor_load_to_lds(g0, g1, z4, z4, z8, 0);
#else
    __builtin_amdgcn_tensor_load_to_lds(g0, g1, z4, z4, 0);
#endif
}
#endif // HAVE_TDM

// Shared weight-scan + accumulate epilogue pieces are inlined manually below.

// ---------------------------------------------------------------------------
// MAIN kernel, non-template and FIRST in the file so the asm snippet opens
// with it: alpha compositing, one thread per pixel, 256-pixel tile per block.
// Fragments/alphas are streamed into LDS by the Tensor Data Mover; weights
// come from a K=8 transmittance scan; features (pre-transposed to ft[P][32])
// are gathered as contiguous 128B columns; outputs stored coalesced and
// non-temporal (write-once; keeps ft resident in L2).
// ---------------------------------------------------------------------------
__global__ __launch_bounds__(256)
void composite_tdm_kernel(const long long* __restrict__ frag,
                          const float* __restrict__ alpha,
                          const float* __restrict__ ft,    // ft[P][32]
                          float* __restrict__ out)
{
    __shared__ long long sh_frag [K_ * TILE_];   // 16 KB
    __shared__ float     sh_alpha[K_ * TILE_];   //  8 KB

    const int t   = threadIdx.x;
    const int n   = blockIdx.x >> 8;             // HW_/TILE_ == 256 tiles per image
    const int hw0 = (blockIdx.x & 255) << 8;

#if HAVE_TDM
    if (t < 32) {
        const uint64_t fbase = (uint64_t)(uintptr_t)frag +
                               ((uint64_t)(n * K_) * HW_ + (uint64_t)hw0) * 8ull;
        const uint64_t abase = (uint64_t)(uintptr_t)alpha +
                               ((uint64_t)(n * K_) * HW_ + (uint64_t)hw0) * 4ull;
        // 2D tiles: 256 contiguous pixels x 8 layers, row stride HW_ elements.
        tdm_load_2d((uint32_t)(uintptr_t)sh_frag,  fbase, /*8B*/3u,
                    TILE_, K_, (uint64_t)HW_, (uint64_t)K_, (uint64_t)HW_);
        tdm_load_2d((uint32_t)(uintptr_t)sh_alpha, abase, /*4B*/2u,
                    TILE_, K_, (uint64_t)HW_, (uint64_t)K_, (uint64_t)HW_);
        __builtin_amdgcn_s_wait_tensorcnt(0);
    }
    __syncthreads();
#else
#pragma unroll
    for (int k = 0; k < K_; ++k) {
        sh_frag [k * TILE_ + t] = frag [(uint64_t)(n * K_ + k) * HW_ + hw0 + t];
        sh_alpha[k * TILE_ + t] = alpha[(uint64_t)(n * K_ + k) * HW_ + hw0 + t];
    }
    __syncthreads();
#endif

    float acc[C_];
#pragma unroll
    for (int c = 0; c < C_; ++c) acc[c] = 0.0f;

    float T = 1.0f;                              // exclusive transmittance
#pragma unroll
    for (int k = 0; k < K_; ++k) {
        const long long f = sh_frag [k * TILE_ + t];
        float           a = sh_alpha[k * TILE_ + t];
        const bool valid  = (f >= 0);
        a = valid ? a : 0.0f;
        const float w = a * T;
        T *= (1.0f - a);
        const uint32_t idx = valid ? (uint32_t)f : 0u;

        const float4* __restrict__ col = (const float4*)(ft + (uint64_t)idx * C_);
#pragma unroll
        for (int c4 = 0; c4 < C_ / 4; ++c4) {    // 8 x global_load_b128, contiguous
            const float4 v = col[c4];
            acc[4 * c4 + 0] = fmaf(w, v.x, acc[4 * c4 + 0]);
            acc[4 * c4 + 1] = fmaf(w, v.y, acc[4 * c4 + 1]);
            acc[4 * c4 + 2] = fmaf(w, v.z, acc[4 * c4 + 2]);
            acc[4 * c4 + 3] = fmaf(w, v.w, acc[4 * c4 + 3]);
        }
    }

    float* __restrict__ op = out + (uint64_t)n * C_ * HW_ + hw0 + t;
#pragma unroll
    for (int c = 0; c < C_; ++c)                 // coalesced across lanes (w-major)
        __builtin_nontemporal_store(acc[c], op + (uint64_t)c * HW_);
}

// ---------------------------------------------------------------------------
// Fallback (workspace too small for the transpose): gather from the original
// features[c][p] layout. No LDS staging needed; loads are plain coalesced.
// ---------------------------------------------------------------------------
__global__ __launch_bounds__(256)
void composite_raw_kernel(const long long* __restrict__ frag,
                          const float* __restrict__ alpha,
                          const float* __restrict__ feat,  // [C][P]
                          float* __restrict__ out, int P)
{
    const int t   = threadIdx.x;
    const int n   = blockIdx.x >> 8;
    const int hw0 = (blockIdx.x & 255) << 8;

    float acc[C_];
#pragma unroll
    for (int c = 0; c < C_; ++c) acc[c] = 0.0f;

    float T = 1.0f;
#pragma unroll
    for (int k = 0; k < K_; ++k) {
        const long long f = frag [(uint64_t)(n * K_ + k) * HW_ + hw0 + t];
        float           a = alpha[(uint64_t)(n * K_ + k) * HW_ + hw0 + t];
        const bool valid  = (f >= 0);
        a = valid ? a : 0.0f;
        const float w = a * T;
        T *= (1.0f - a);
        const uint32_t idx = valid ? (uint32_t)f : 0u;
#pragma unroll
        for (int c = 0; c < C_; ++c)
            acc[c] = fmaf(w, feat[(uint64_t)c * P + idx], acc[c]);
    }

    float* __restrict__ op = out + (uint64_t)n * C_ * HW_ + hw0 + t;
#pragma unroll
    for (int c = 0; c < C_; ++c)
        __builtin_nontemporal_store(acc[c], op + (uint64_t)c * HW_);
}

// ---------------------------------------------------------------------------
// Transpose features (C=32 x P) -> ft (P x 32): each fragment gather becomes
// one contiguous 128B read. Input read non-temporally (read exactly once);
// output stored regular-temporal so ft stays L2-resident.
// ---------------------------------------------------------------------------
__global__ __launch_bounds__(256)
void transpose_features(const float* __restrict__ feat, float* __restrict__ ft, int P)
{
    __shared__ float tile[32][33];               // +1 avoids LDS bank conflicts
    const int p0 = blockIdx.x * 32;
    const int x  = threadIdx.x;                  // 0..31
    for (int c = threadIdx.y; c < 32; c += 8) {
        const int p = p0 + x;
        tile[c][x] = (p < P) ? __builtin_nontemporal_load(feat + (uint64_t)c * P + p)
                             : 0.0f;
    }
    __syncthreads();
    for (int r = threadIdx.y; r < 32; r += 8) {
        const int p = p0 + r;
        if (p < P) ft[(uint64_t)p * 32 + x] = tile[x][r];
    }
}

// ---------------------------------------------------------------------------
extern "C" void kernel_launch(void* const* d_in, const int* in_sizes, int n_in,
                              void* d_out, int out_size, void* d_ws, size_t ws_size,
                              hipStream_t stream)
{
    const long long* frag  = (const long long*)d_in[0];
    const float*     alpha = (const float*)d_in[1];
    const float*     feat  = (const float*)d_in[2];
    float*           out   = (float*)d_out;
    const int        P     = in_sizes[2] / C_;

    const int blocks = N_ * (HW_ / TILE_);       // 2048

    if (ws_size >= (size_t)P * C_ * sizeof(float)) {
        float* ft = (float*)d_ws;
        transpose_features<<<(P + 31) / 32, dim3(32, 8), 0, stream>>>(feat, ft, P);
        composite_tdm_kernel<<<blocks, TILE_, 0, stream>>>(frag, alpha, ft, out);
    } else {
        composite_raw_kernel<<<blocks, TILE_, 0, stream>>>(frag, alpha, feat, out, P);
    }
}